// ApsPool_6382321401913
// MI455X (gfx1250) — compile-verified
//
#include <hip/hip_runtime.h>

typedef __attribute__((ext_vector_type(2))) float v2f;
typedef __attribute__((ext_vector_type(8))) float v8f;

#define B_SZ 32
#define T_SZ 64
#define F_SZ 64
#define C_SZ 128
#define T_CH 16     // t rows per block
#define F_CH 16     // f cols per block
#define XF   25     // staged x f-extent: fabs in [fc0-4, fc0+20]
#define XS   128    // x row stride (floats), packed for b128 async copies
#define PF   24     // P row f-extent: j in [fc0-4, fc0+19]
#define PS   144    // padded P stride: half-wave B reads hit disjoint banks
#define HS   130    // padded H stride: M=r vs M=r+8 stores hit disjoint banks

// ---------------- CDNA5 async global->LDS helpers ----------------

__device__ __forceinline__ void wait_async0() {
#if __has_builtin(__builtin_amdgcn_s_wait_asynccnt)
    __builtin_amdgcn_s_wait_asynccnt(0);
#else
    asm volatile("s_wait_asynccnt 0x0" ::: "memory");
#endif
}

// Stage x[row, fc0-4 .. fc0+20, 0..127] into ldsrow[pf*XS + c] with b128 async copies.
__device__ __forceinline__ void async_x_row(const float* __restrict__ xb, int row,
                                            float* ldsrow, int fc0, int tid) {
    const float* gr = xb + (size_t)row * F_SZ * C_SZ;
#pragma unroll
    for (int it = 0; it < 4; ++it) {
        const int idx = tid + it * 256;              // b128 chunk id
        if (idx < XF * (C_SZ / 4)) {
            const int pf   = idx >> 5;               // 32 quads per f line
            const int cq   = (idx & 31) << 2;
            const int fabs = fc0 - 4 + pf;
            if ((unsigned)fabs < (unsigned)F_SZ) {
                const float* g = gr + fabs * C_SZ + cq;
                const uint32_t l = (uint32_t)(uintptr_t)(ldsrow + pf * XS + cq);
                asm volatile("global_load_async_to_lds_b128 %0, %1, off"
                             :: "v"(l), "v"(g) : "memory");
            }
        }
    }
}

// ---------------- max-pool stage (LDS x-ring -> LDS P row) ----------------

// P[j] = 2x2 stride-1 SAME max-pool (window [t..t+1]x[j..j+1], clamped top),
// stored as 0 outside [0,F) to realize the blur's zero padding.
__device__ __forceinline__ void compute_P_row(const float* __restrict__ Xc,
                                              const float* __restrict__ Xn, bool haveNext,
                                              float* __restrict__ Prow, int fc0, int tid) {
    const int cc = tid & 127;
    const int f0 = (tid >> 7) * (PF / 2);            // 0 or 12
#pragma unroll
    for (int i = 0; i < PF / 2; ++i) {
        const int pf = f0 + i;
        const int j  = fc0 - 4 + pf;
        float v = 0.0f;
        if ((unsigned)j < (unsigned)F_SZ) {
            v = Xc[pf * XS + cc];
            const bool fnext = (j + 1 < F_SZ);
            if (fnext) v = fmaxf(v, Xc[(pf + 1) * XS + cc]);
            if (haveNext) {
                v = fmaxf(v, Xn[pf * XS + cc]);
                if (fnext) v = fmaxf(v, Xn[(pf + 1) * XS + cc]);
            }
        }
        Prow[pf * PS + cc] = v;
    }
}

// ---------------- horizontal blur via V_WMMA_F32_16X16X4_F32 ----------------
// D(16f x 16c) = sum_k A(16f x 4k banded [1,2,1]/4) * B(4k x 16c pooled data).
// A 16x4: lanes 0-15 rows, VGPR0/1 = K0/K1; lanes 16-31 K2/K3.
// B 4x16: VGPR0 lanes0-15 K=0, lanes16-31 K=2; VGPR1 K=1/K=3.
// D 16x16: VGPR r, lanes0-15 -> (M=r, N=lane); lanes16-31 -> (M=r+8, N=lane-16).

__device__ __forceinline__ float wgt3(int d) {
    if (d == 0) return 0.50f;
    if (d == 1 || d == -1) return 0.25f;
    return 0.0f;
}

__device__ __forceinline__ void build_A(v2f* Aw, int lane) {
    const int mrow  = lane & 15;
    const int khalf = (lane >> 4) << 1;
#pragma unroll
    for (int ch = 0; ch < 6; ++ch) {
        const int d0 = -4 + (ch << 2) + khalf;       // (k - m) offset base
        Aw[ch].x = wgt3(d0 - mrow);
        Aw[ch].y = wgt3(d0 + 1 - mrow);
    }
}

__device__ __forceinline__ void compute_H_row(const float* __restrict__ Prow,
                                              float* __restrict__ Hrow,
                                              const v2f* Aw, int lane, int wave) {
    const int mrow  = lane & 15;
    const int khalf = (lane >> 4) << 1;
    const int c     = (wave << 4) + mrow;            // this lane's channel (B/D column)
    v8f acc = {};
#pragma unroll
    for (int ch = 0; ch < 6; ++ch) {
        const int pf = (ch << 2) + khalf;            // K index into P row
        v2f bv;
        bv.x = Prow[pf * PS + c];
        bv.y = Prow[(pf + 1) * PS + c];
        acc = __builtin_amdgcn_wmma_f32_16x16x4_f32(
            false, Aw[ch], false, bv, (short)0, acc, false, false);
    }
    const int fbase = (lane >> 4) << 3;
#pragma unroll
    for (int r = 0; r < 8; ++r)
        Hrow[(fbase + r) * HS + c] = acc[r];
}

// ---------------- Pass 1: per-(batch, phase) sum of squares ----------------

__global__ __launch_bounds__(256) void aps_norm_kernel(const float* __restrict__ x,
                                                       float* __restrict__ norms) {
    __shared__ float Xring[2][XF * XS];   // 25.6 KB raw x rows (async-filled)
    __shared__ float Psh[PF * PS];        // 13.8 KB pooled row (reused as reduce scratch)
    __shared__ float Hring[3][F_CH * HS]; // 24.9 KB H-blurred row ring

    const int b   = blockIdx.y;
    const int t0  = (blockIdx.x >> 2) * T_CH;
    const int fc0 = (blockIdx.x & 3) * F_CH;
    const float* xb = x + (size_t)b * T_SZ * F_SZ * C_SZ;

    const int tid = threadIdx.x, lane = tid & 31, wave = tid >> 5;
    v2f Aw[6];
    build_A(Aw, lane);

    const int fi    = tid >> 4;                      // 0..15 (f within chunk)
    const int cst   = (tid & 15) << 3;               // 8 channels
    const int fpar2 = ((fc0 + fi) & 1) << 1;
    float accE = 0.f, accO = 0.f;                    // t-even / t-odd phase sums

    // rotating H-ring pointers: pA = H[t-2], pB = H[t-1], pC = write target (H[t])
    float* pA = Hring[0];
    float* pB = Hring[1];
    float* pC = Hring[2];

    const int lastXrow = (t0 + T_CH + 1 < T_SZ) ? (t0 + T_CH + 1) : (T_SZ - 1);
    if (t0 - 1 >= 0) async_x_row(xb, t0 - 1, Xring[(t0 - 1) & 1], fc0, tid);

    for (int tr = t0 - 1; tr <= t0 + T_CH; ++tr) {
        const int nrow = tr + 1;
        if (nrow <= lastXrow) async_x_row(xb, nrow, Xring[nrow & 1], fc0, tid);
        wait_async0();
        __syncthreads();

        if ((unsigned)tr < (unsigned)T_SZ) {
            compute_P_row(Xring[tr & 1], Xring[(tr + 1) & 1], tr + 1 < T_SZ, Psh, fc0, tid);
            __syncthreads();
            compute_H_row(Psh, pC, Aw, lane, wave);
        } else {
            for (int i = tid; i < F_CH * HS; i += 256) pC[i] = 0.f;
            __syncthreads();
        }
        __syncthreads();

        const int t = tr - 1;
        if (t >= t0) {
            float s = 0.f;
#pragma unroll
            for (int cc = 0; cc < 8; ++cc) {
                const int o  = fi * HS + cst + cc;
                const float bl = 0.25f * (pA[o] + 2.0f * pB[o] + pC[o]);
                s += bl * bl;
            }
            if (t & 1) accO += s; else accE += s;
        }
        __syncthreads();

        float* tmp = pA; pA = pB; pB = pC; pC = tmp; // rotate ring
    }

    // block reduce (reuse Psh), then one atomic per (batch, phase)
    float* red = Psh;
#pragma unroll
    for (int k = 0; k < 4; ++k)
        red[tid * 4 + k] = (k == fpar2) ? accE : (k == fpar2 + 1) ? accO : 0.f;
    __syncthreads();
    if (tid < 4) {
        float s = 0.f;
        for (int i = 0; i < 256; ++i) s += red[i * 4 + tid];
        atomicAdd(&norms[b * 4 + tid], s);
    }
}

// ---------------- Pass 2: argmax phase, gather selected component ----------------

__global__ __launch_bounds__(256) void aps_gather_kernel(const float* __restrict__ x,
                                                         const float* __restrict__ norms,
                                                         float* __restrict__ out) {
    __shared__ float Xring[2][XF * XS];
    __shared__ float Psh[PF * PS];
    __shared__ float Hring[3][F_CH * HS];

    const int b   = blockIdx.y;
    const int t0  = (blockIdx.x >> 2) * T_CH;
    const int fc0 = (blockIdx.x & 3) * F_CH;
    const float* xb = x + (size_t)b * T_SZ * F_SZ * C_SZ;

    // first-max argmax over 4 candidate norms (matches jnp.argmax)
    const float n0 = norms[b * 4 + 0], n1 = norms[b * 4 + 1];
    const float n2 = norms[b * 4 + 2], n3 = norms[b * 4 + 3];
    int idx = 0; float best = n0;
    if (n1 > best) { best = n1; idx = 1; }
    if (n2 > best) { best = n2; idx = 2; }
    if (n3 > best) { best = n3; idx = 3; }
    const int pt = idx & 1;                          // t parity
    const int pf = (idx >> 1) & 1;                   // f parity

    const int tid = threadIdx.x, lane = tid & 31, wave = tid >> 5;
    v2f Aw[6];
    build_A(Aw, lane);

    // output mapping: 8 selected f x 128 c over 256 threads (4 c each)
    const int fsel  = tid >> 5;                      // 0..7
    const int f_rel = (fsel << 1) + pf;
    const int f_abs = fc0 + f_rel;
    const int cst   = (tid & 31) << 2;               // 4 channels

    float* pA = Hring[0];
    float* pB = Hring[1];
    float* pC = Hring[2];

    const int lastXrow = (t0 + T_CH + 1 < T_SZ) ? (t0 + T_CH + 1) : (T_SZ - 1);
    if (t0 - 1 >= 0) async_x_row(xb, t0 - 1, Xring[(t0 - 1) & 1], fc0, tid);

    for (int tr = t0 - 1; tr <= t0 + T_CH; ++tr) {
        const int nrow = tr + 1;
        if (nrow <= lastXrow) async_x_row(xb, nrow, Xring[nrow & 1], fc0, tid);
        wait_async0();
        __syncthreads();

        if ((unsigned)tr < (unsigned)T_SZ) {
            compute_P_row(Xring[tr & 1], Xring[(tr + 1) & 1], tr + 1 < T_SZ, Psh, fc0, tid);
            __syncthreads();
            compute_H_row(Psh, pC, Aw, lane, wave);
        } else {
            for (int i = tid; i < F_CH * HS; i += 256) pC[i] = 0.f;
            __syncthreads();
        }
        __syncthreads();

        const int t = tr - 1;
        if (t >= t0 && (t & 1) == pt) {
            float* op = out + (((size_t)b * (T_SZ / 2) + (t >> 1)) * (F_SZ / 2) + (f_abs >> 1)) * C_SZ + cst;
#pragma unroll
            for (int cc = 0; cc < 4; ++cc) {
                const int o = f_rel * HS + cst + cc;
                op[cc] = 0.25f * (pA[o] + 2.0f * pB[o] + pC[o]);
            }
        }
        __syncthreads();

        float* tmp = pA; pA = pB; pB = pC; pC = tmp; // rotate ring
    }
}

extern "C" void kernel_launch(void* const* d_in, const int* in_sizes, int n_in,
                              void* d_out, int out_size, void* d_ws, size_t ws_size,
                              hipStream_t stream) {
    const float* x = (const float*)d_in[0];
    // d_in[1] is the fixed binomial blur kernel; coefficients baked in ([1,2,1]/4 separable).
    float* out   = (float*)d_out;
    float* norms = (float*)d_ws;                     // 32 batches x 4 phases

    hipMemsetAsync(norms, 0, B_SZ * 4 * sizeof(float), stream);

    dim3 grid(16, B_SZ);                             // x: 4 t-chunks * 4 f-chunks, y: batch
    aps_norm_kernel<<<grid, 256, 0, stream>>>(x, norms);
    aps_gather_kernel<<<grid, 256, 0, stream>>>(x, norms, out);
}